// TransitionLayerNoTime_3332894621738
// MI455X (gfx1250) — compile-verified
//
#include <hip/hip_runtime.h>
#include <math.h>

typedef __attribute__((ext_vector_type(2))) float v2f;
typedef __attribute__((ext_vector_type(8))) float v8f;

#define CODE 18000
#define GRP  256
#define HID  256
#define ATT  128
#define G1   6000
#define G23  12000

__device__ __forceinline__ v8f wmma4(v2f a, v2f b, v8f c) {
    // V_WMMA_F32_16X16X4_F32 : D = A(16x4) x B(4x16) + C(16x16), fp32
    return __builtin_amdgcn_wmma_f32_16x16x4_f32(
        false, a, false, b, (short)0, c, false, false);
}

// CDNA5 async global->LDS copy (16B per lane), tracked by ASYNCcnt.
// LDS byte address = low 32 bits of the generic pointer.
__device__ __forceinline__ void async_copy_b128(const float* g, float* l) {
    unsigned lds = (unsigned)(uintptr_t)l;
    asm volatile("global_load_async_to_lds_b128 %0, %1, off"
                 :: "v"(lds), "v"(g) : "memory");
}
__device__ __forceinline__ void wait_async0() {
    asm volatile("s_wait_asynccnt 0x0" ::: "memory");
}

// ---------------------------------------------------------------- init
__global__ void init_kernel(float* __restrict__ out) {
    size_t i = (size_t)blockIdx.x * 256 + threadIdx.x;
    if (i < 256) out[i] = -INFINITY;
    else if (i < (size_t)256 + (size_t)CODE * HID) out[i] = 0.0f;
}

// ---------------------------------------------------------------- Q/K/V projection
// Block = 256 threads (8 waves), handles 16 gathered rows of q_in / v_in.
// 32 column tiles of 16: [0,8)=Q via wq, [8,16)=K via wk, [16,32)=V via wv.
__global__ __launch_bounds__(256) void qkv_kernel(
    const float* __restrict__ co, const float* __restrict__ no_e,
    const float* __restrict__ unrel,
    const float* __restrict__ wq, const float* __restrict__ bq,
    const float* __restrict__ wk, const float* __restrict__ bk,
    const float* __restrict__ wv, const float* __restrict__ bv,
    const int* __restrict__ m2, const int* __restrict__ m3,
    float* __restrict__ Qo, float* __restrict__ Ko, float* __restrict__ Vo) {
    __shared__ float Qin[16 * 260];   // padded stride 260 (bank-conflict free A reads)
    __shared__ float Vin[16 * 260];
    const int tid  = threadIdx.x;
    const int row0 = blockIdx.x * 16;

    // cooperative gather of 16 rows (float4 per thread x4)
    for (int i = 0; i < 4; ++i) {
        int j = tid + i * 256;          // 1024 float4 total per buffer
        int r = j >> 6, off = (j & 63) << 2;
        int gr = row0 + r;
        const float *qsrc, *vsrc;
        if (gr < G1) { int idx = m2[gr];       qsrc = no_e  + (long)idx * GRP; vsrc = co + (long)idx * GRP; }
        else         { int idx = m3[gr - G1];  qsrc = unrel + (long)idx * GRP; vsrc = co + (long)idx * GRP; }
        *(float4*)&Qin[r * 260 + off] = *(const float4*)(qsrc + off);
        *(float4*)&Vin[r * 260 + off] = *(const float4*)(vsrc + off);
    }
    __syncthreads();

    const int wave = tid >> 5, lane = tid & 31;
    const int n = lane & 15, h = lane >> 4;

    for (int tt = 0; tt < 4; ++tt) {
        int tile = wave * 4 + tt;
        const float *A, *W, *B; float* O; int ldw, col0;
        if (tile < 8)       { A = Qin; W = wq; B = bq; O = Qo; ldw = ATT; col0 = tile * 16; }
        else if (tile < 16) { A = Vin; W = wk; B = bk; O = Ko; ldw = ATT; col0 = (tile - 8) * 16; }
        else                { A = Vin; W = wv; B = bv; O = Vo; ldw = HID; col0 = (tile - 16) * 16; }
        v8f acc = {};
        for (int ks = 0; ks < 64; ++ks) {
            int k = ks * 4 + 2 * h;                           // lane-half selects K pair
            v2f a = *(const v2f*)&A[n * 260 + k];             // A(M=n, K=k..k+1)
            const float* wp = W + (long)k * ldw + col0 + n;   // B(kk, n)
            v2f b; b[0] = wp[0]; b[1] = wp[ldw];
            acc = wmma4(a, b, acc);
        }
        float bias = B[col0 + n];
        #pragma unroll
        for (int v = 0; v < 8; ++v) {
            int M = v + 8 * h;                                 // C/D layout
            O[(long)(row0 + M) * ldw + col0 + n] = acc[v] + bias;
        }
    }
}

// ---------------------------------------------------------------- GRU (m1 group)
__global__ __launch_bounds__(256) void gru_kernel(
    const float* __restrict__ co, const float* __restrict__ hs,
    const float* __restrict__ w_ih, const float* __restrict__ w_hh,
    const float* __restrict__ b_ih, const float* __restrict__ b_hh,
    const int* __restrict__ m1,
    float* __restrict__ out, float* __restrict__ h_new) {
    __shared__ float X[16 * 260];
    __shared__ float H[16 * 260];
    const int tid  = threadIdx.x;
    const int row0 = blockIdx.x * 16;

    for (int i = 0; i < 4; ++i) {
        int j = tid + i * 256;
        int r = j >> 6, off = (j & 63) << 2;
        int idx = m1[row0 + r];
        *(float4*)&X[r * 260 + off] = *(const float4*)(co + (long)idx * GRP + off);
        *(float4*)&H[r * 260 + off] = *(const float4*)(hs + (long)idx * HID + off);
    }
    __syncthreads();

    const int wave = tid >> 5, lane = tid & 31;
    const int n = lane & 15, h = lane >> 4;

    for (int tt = 0; tt < 2; ++tt) {
        int col0 = (wave * 2 + tt) * 16;
        v8f ir = {}, iz = {}, in_ = {}, hr = {}, hz = {}, hn = {};
        for (int ks = 0; ks < 64; ++ks) {
            int k = ks * 4 + 2 * h;
            v2f ax = *(const v2f*)&X[n * 260 + k];
            v2f ah = *(const v2f*)&H[n * 260 + k];
            const float* wi = w_ih + (long)k * 768 + col0 + n;
            const float* wh = w_hh + (long)k * 768 + col0 + n;
            v2f b;
            b[0] = wi[0];         b[1] = wi[768];         ir  = wmma4(ax, b, ir);
            b[0] = wi[256];       b[1] = wi[768 + 256];   iz  = wmma4(ax, b, iz);
            b[0] = wi[512];       b[1] = wi[768 + 512];   in_ = wmma4(ax, b, in_);
            b[0] = wh[0];         b[1] = wh[768];         hr  = wmma4(ah, b, hr);
            b[0] = wh[256];       b[1] = wh[768 + 256];   hz  = wmma4(ah, b, hz);
            b[0] = wh[512];       b[1] = wh[768 + 512];   hn  = wmma4(ah, b, hn);
        }
        float bir = b_ih[col0 + n], biz = b_ih[256 + col0 + n], bin = b_ih[512 + col0 + n];
        float bhr = b_hh[col0 + n], bhz = b_hh[256 + col0 + n], bhn = b_hh[512 + col0 + n];
        #pragma unroll
        for (int v = 0; v < 8; ++v) {
            int M = v + 8 * h;
            float r  = 1.0f / (1.0f + __expf(-(ir[v] + bir + hr[v] + bhr)));
            float z  = 1.0f / (1.0f + __expf(-(iz[v] + biz + hz[v] + bhz)));
            float nn = tanhf(in_[v] + bin + r * (hn[v] + bhn));
            float hv = H[M * 260 + col0 + n];
            float res = (1.0f - z) * nn + z * hv;
            int dest = m1[row0 + M];
            h_new[(long)dest * HID + col0 + n] = res;
            atomicMax(&out[col0 + n], res);
        }
    }
}

// ---------------------------------------------------------------- flash attention (m2+m3)
// Block = 8 waves; each wave owns a 16-row query tile; block cooperatively
// stages 16-key K/V chunks in double-buffered LDS via ASYNC global->LDS DMA,
// overlapping the copy of chunk it+1 with the 96 WMMAs of chunk it.
__global__ __launch_bounds__(256) void attn_kernel(
    const float* __restrict__ Q, const float* __restrict__ K,
    const float* __restrict__ V,
    const int* __restrict__ m2, const int* __restrict__ m3,
    float* __restrict__ out, float* __restrict__ h_new) {
    __shared__ float Kb[2][16 * 132];       // padded stride
    __shared__ float Vb[2][16 * 256];
    __shared__ float Pb[8 * 16 * 17];       // per-wave P scratch, padded
    const int tid  = threadIdx.x;
    const int wave = tid >> 5, lane = tid & 31;
    const int n = lane & 15, h = lane >> 4;
    const int qt = blockIdx.x * 8 + wave;
    const bool active = (qt * 16 < G23);    // wave-uniform -> EXEC all-ones in WMMA
    const int NIT = G23 / 16;

    v2f qa[32];
    if (active) {
        const float* qp = Q + (long)(qt * 16 + n) * ATT;
        const float scale = 0.08838834764831845f;   // 1/sqrt(128)
        for (int ks = 0; ks < 32; ++ks) {
            int k = ks * 4 + 2 * h;
            qa[ks][0] = qp[k] * scale;
            qa[ks][1] = qp[k + 1] * scale;
        }
    }
    v8f o[16];
    v8f zero = {};
    #pragma unroll
    for (int t = 0; t < 16; ++t) o[t] = zero;
    float m_[8], l_[8];
    #pragma unroll
    for (int v = 0; v < 8; ++v) { m_[v] = -INFINITY; l_[v] = 0.0f; }
    float* Pw = &Pb[wave * 16 * 17];

    // prologue: async-stage chunk 0 into buffer 0
    {
        for (int i = 0; i < 2; ++i) {                 // 512 x b128 of K
            int j = tid + i * 256;
            int r = j >> 5, off = (j & 31) << 2;
            async_copy_b128(K + (long)r * ATT + off, &Kb[0][r * 132 + off]);
        }
        for (int i = 0; i < 4; ++i) {                 // 1024 x b128 of V
            int j = tid + i * 256;
            int r = j >> 6, off = (j & 63) << 2;
            async_copy_b128(V + (long)r * HID + off, &Vb[0][r * 256 + off]);
        }
    }
    wait_async0();
    __syncthreads();

    for (int it = 0; it < NIT; ++it) {
        const int cur = it & 1;
        // issue async copies for the next chunk while computing this one
        if (it + 1 < NIT) {
            const int key0 = (it + 1) * 16;
            const int nxt = cur ^ 1;
            for (int i = 0; i < 2; ++i) {
                int j = tid + i * 256;
                int r = j >> 5, off = (j & 31) << 2;
                async_copy_b128(K + (long)(key0 + r) * ATT + off, &Kb[nxt][r * 132 + off]);
            }
            for (int i = 0; i < 4; ++i) {
                int j = tid + i * 256;
                int r = j >> 6, off = (j & 63) << 2;
                async_copy_b128(V + (long)(key0 + r) * HID + off, &Vb[nxt][r * 256 + off]);
            }
        }

        if (active) {
            v8f s = {};
            for (int ks = 0; ks < 32; ++ks) {             // S = Qs K^T
                int k = ks * 4 + 2 * h;
                v2f b = *(const v2f*)&Kb[cur][n * 132 + k];
                s = wmma4(qa[ks], b, s);
            }
            float scl[8];
            #pragma unroll
            for (int v = 0; v < 8; ++v) {                 // online softmax per row
                float rm = s[v];
                rm = fmaxf(rm, __shfl_xor(rm, 1, 32));
                rm = fmaxf(rm, __shfl_xor(rm, 2, 32));
                rm = fmaxf(rm, __shfl_xor(rm, 4, 32));
                rm = fmaxf(rm, __shfl_xor(rm, 8, 32));
                float nm = fmaxf(m_[v], rm);
                scl[v] = __expf(m_[v] - nm);
                float p = __expf(s[v] - nm);
                s[v] = p;
                float rs = p;
                rs += __shfl_xor(rs, 1, 32);
                rs += __shfl_xor(rs, 2, 32);
                rs += __shfl_xor(rs, 4, 32);
                rs += __shfl_xor(rs, 8, 32);
                l_[v] = l_[v] * scl[v] + rs;
                m_[v] = nm;
            }
            #pragma unroll
            for (int t = 0; t < 16; ++t)
                #pragma unroll
                for (int v = 0; v < 8; ++v) o[t][v] *= scl[v];

            // C-layout -> A-layout via per-wave LDS scratch
            #pragma unroll
            for (int v = 0; v < 8; ++v) Pw[(v + 8 * h) * 17 + n] = s[v];
            __builtin_amdgcn_wave_barrier();
            v2f pa[4];
            #pragma unroll
            for (int ks = 0; ks < 4; ++ks) {
                int k = ks * 4 + 2 * h;
                pa[ks][0] = Pw[n * 17 + k];
                pa[ks][1] = Pw[n * 17 + k + 1];
            }
            for (int t = 0; t < 16; ++t) {                // O += P V
                #pragma unroll
                for (int ks = 0; ks < 4; ++ks) {
                    int kk = ks * 4 + 2 * h;
                    v2f b;
                    b[0] = Vb[cur][kk * 256 + t * 16 + n];
                    b[1] = Vb[cur][(kk + 1) * 256 + t * 16 + n];
                    o[t] = wmma4(pa[ks], b, o[t]);
                }
            }
        }
        wait_async0();          // next buffer complete (hidden behind compute)
        __syncthreads();        // all waves done reading cur before it is re-filled
    }

    if (active) {
        #pragma unroll
        for (int v = 0; v < 8; ++v) {
            int M = v + 8 * h;
            int grow = qt * 16 + M;
            int dest = (grow < G1) ? m2[grow] : m3[grow - G1];
            float inv = 1.0f / l_[v];
            #pragma unroll
            for (int t = 0; t < 16; ++t) {
                float val = tanhf(o[t][v] * inv);
                h_new[(long)dest * HID + t * 16 + n] = val;
                atomicMax(&out[t * 16 + n], val);
            }
        }
    }
}

// ---------------------------------------------------------------- launch
extern "C" void kernel_launch(void* const* d_in, const int* in_sizes, int n_in,
                              void* d_out, int out_size, void* d_ws, size_t ws_size,
                              hipStream_t stream) {
    const float* co   = (const float*)d_in[0];
    const float* no_e = (const float*)d_in[1];
    const float* un   = (const float*)d_in[2];
    const float* hs   = (const float*)d_in[3];
    const float* w_ih = (const float*)d_in[4];
    const float* w_hh = (const float*)d_in[5];
    const float* b_ih = (const float*)d_in[6];
    const float* b_hh = (const float*)d_in[7];
    const float* wq   = (const float*)d_in[8];
    const float* bq   = (const float*)d_in[9];
    const float* wk   = (const float*)d_in[10];
    const float* bk   = (const float*)d_in[11];
    const float* wv   = (const float*)d_in[12];
    const float* bv   = (const float*)d_in[13];
    const int*   m1   = (const int*)d_in[14];
    const int*   m2   = (const int*)d_in[15];
    const int*   m3   = (const int*)d_in[16];

    float* out   = (float*)d_out;        // [256] column max
    float* h_new = out + 256;            // [18000, 256]

    float* Qw = (float*)d_ws;            // [12000, 128]
    float* Kw = Qw + (size_t)G23 * ATT;  // [12000, 128]
    float* Vw = Kw + (size_t)G23 * ATT;  // [12000, 256]

    size_t tot = 256 + (size_t)CODE * HID;
    init_kernel<<<(tot + 255) / 256, 256, 0, stream>>>(out);
    qkv_kernel<<<G23 / 16, 256, 0, stream>>>(co, no_e, un, wq, bq, wk, bk, wv, bv,
                                             m2, m3, Qw, Kw, Vw);
    gru_kernel<<<G1 / 16, 256, 0, stream>>>(co, hs, w_ih, w_hh, b_ih, b_hh,
                                            m1, out, h_new);
    attn_kernel<<<(G23 / 16 + 7) / 8, 256, 0, stream>>>(Qw, Kw, Vw, m2, m3, out, h_new);
}